// PillarFeatureEnconder_5961414607099
// MI455X (gfx1250) — compile-verified
//
#include <hip/hip_runtime.h>

typedef float v2f __attribute__((ext_vector_type(2)));
typedef float v4f __attribute__((ext_vector_type(4)));
typedef float v8f __attribute__((ext_vector_type(8)));

#define NX_   432
#define NY_   496
#define GRID_ (NX_ * NY_)        // 214272
#define NPT_  32
#define VX_   0.16f
#define VY_   0.16f
#define VZ_   4.0f
#define XOFF_ (VX_ * 0.5f + 0.0f)
#define YOFF_ (VY_ * 0.5f + (-39.68f))
#define ZOFF_ (VZ_ * 0.5f + (-3.0f))
#define EPS_  1e-3f

// ---------------------------------------------------------------------------
// Shared feature computation (must match between stats and main pass)
// feat = [x,y,z,w, x-mx,y-my,z-mz, x-cx,y-cy,z-cz] * mask
// ---------------------------------------------------------------------------
__device__ __forceinline__ void compute_feat(float vx, float vy, float vz, float vw,
                                             float sx, float sy, float sz, float npf,
                                             int cz, int cy, int cx, bool valid,
                                             float* f) {
  float mx = sx / npf, my = sy / npf, mz = sz / npf;
  float ccx = (float)cx * VX_ + XOFF_;
  float ccy = (float)cy * VY_ + YOFF_;
  float ccz = (float)cz * VZ_ + ZOFF_;
  f[0] = vx; f[1] = vy; f[2] = vz; f[3] = vw;
  f[4] = vx - mx;  f[5] = vy - my;  f[6] = vz - mz;
  f[7] = vx - ccx; f[8] = vy - ccy; f[9] = vz - ccz;
  float m = valid ? 1.0f : 0.0f;
#pragma unroll
  for (int i = 0; i < 10; ++i) f[i] *= m;
}

__device__ __forceinline__ float wave_sum(float v) {
#pragma unroll
  for (int off = 16; off >= 1; off >>= 1) v += __shfl_xor(v, off, 32);
  return v;
}

// ---------------------------------------------------------------------------
// Kernel 1: zero-fill the BEV output with non-temporal b128 stores
// ---------------------------------------------------------------------------
__global__ __launch_bounds__(256) void zero_out_k(float* __restrict__ o, long n) {
  long n4 = n >> 2;
  v4f z = {0.f, 0.f, 0.f, 0.f};
  long stride = (long)gridDim.x * blockDim.x;
  for (long i = (long)blockIdx.x * blockDim.x + threadIdx.x; i < n4; i += stride)
    __builtin_nontemporal_store(z, ((v4f*)o) + i);
  if (blockIdx.x == 0 && threadIdx.x == 0)
    for (long i = n4 << 2; i < n; ++i) o[i] = 0.f;
}

// ---------------------------------------------------------------------------
// Kernel 2: zero the 65 stat accumulators in workspace
// ws[0..10)   : S  (sum of feat)
// ws[16..71)  : M  (upper-triangular 10x10 second moment, 55 entries)
// ws[128..192): a_c (scale)   ws[192..256): b_c (shift)
// ---------------------------------------------------------------------------
__global__ void zero_ws_k(float* __restrict__ ws) {
  int i = threadIdx.x;
  if (i < 128) ws[i] = 0.f;
}

// ---------------------------------------------------------------------------
// Kernel 3: accumulate S and M over all P*NPT masked feature rows.
// One pillar per wave per iteration; lane == point index.
// ---------------------------------------------------------------------------
__global__ __launch_bounds__(256) void pillar_stats(const float4* __restrict__ voxels,
                                                    const int* __restrict__ nump,
                                                    const int4* __restrict__ coords,
                                                    float* __restrict__ ws, int P) {
  int lane = threadIdx.x & 31;
  int w = (int)((blockIdx.x * blockDim.x + threadIdx.x) >> 5);
  int nw = (int)((gridDim.x * blockDim.x) >> 5);

  float S[10];
  float M[55];
#pragma unroll
  for (int i = 0; i < 10; ++i) S[i] = 0.f;
#pragma unroll
  for (int t = 0; t < 55; ++t) M[t] = 0.f;

  for (int p = w; p < P; p += nw) {
    float4 v = voxels[(long)p * NPT_ + lane];
    float sx = wave_sum(v.x);
    float sy = wave_sum(v.y);
    float sz = wave_sum(v.z);
    int np = nump[p];
    int4 c = coords[p];                 // (b, z, y, x)
    float f[10];
    compute_feat(v.x, v.y, v.z, v.w, sx, sy, sz, (float)np,
                 c.y, c.z, c.w, lane < np, f);
    int t = 0;
#pragma unroll
    for (int i = 0; i < 10; ++i) {
      S[i] += f[i];
#pragma unroll
      for (int j = i; j < 10; ++j) M[t++] += f[i] * f[j];
    }
  }

  // wave-level reduction, then one atomic per value from lane 0
#pragma unroll
  for (int i = 0; i < 10; ++i) S[i] = wave_sum(S[i]);
#pragma unroll
  for (int t = 0; t < 55; ++t) M[t] = wave_sum(M[t]);
  if (lane == 0) {
#pragma unroll
    for (int i = 0; i < 10; ++i) atomicAdd(&ws[i], S[i]);
#pragma unroll
    for (int t = 0; t < 55; ++t) atomicAdd(&ws[16 + t], M[t]);
  }
}

// ---------------------------------------------------------------------------
// Kernel 4: per-channel BN scale/shift from S, M:
//   mean_c = (W_c . S)/count ;  E[x^2]_c = W_c M W_c^T / count
// ---------------------------------------------------------------------------
__global__ void pillar_finalize(const float* __restrict__ W,
                                const float* __restrict__ gamma,
                                const float* __restrict__ beta,
                                float* __restrict__ ws, float inv_count) {
  int c = threadIdx.x;
  if (c >= 64) return;
  float wr[10];
#pragma unroll
  for (int i = 0; i < 10; ++i) wr[i] = W[c * 10 + i];
  float mean = 0.f;
#pragma unroll
  for (int i = 0; i < 10; ++i) mean += wr[i] * ws[i];
  mean *= inv_count;
  float ex2 = 0.f;
  int t = 0;
#pragma unroll
  for (int i = 0; i < 10; ++i)
#pragma unroll
    for (int j = i; j < 10; ++j, ++t) {
      float coeff = (i == j) ? 1.f : 2.f;
      ex2 += coeff * wr[i] * wr[j] * ws[16 + t];
    }
  ex2 *= inv_count;
  float var = ex2 - mean * mean;
  float inv = rsqrtf(var + EPS_);
  float a = gamma[c] * inv;
  float b = beta[c] - mean * a;
  ws[128 + c] = a;
  ws[192 + c] = b;
}

// ---------------------------------------------------------------------------
// Kernel 5: main pass. One pillar per wave (8 waves / block, exact grid).
// feat (32x12 f32, zero-padded K) staged in LDS; W staged once per block as
// zero-padded 64x12 in LDS (kills divergent predicated global loads).
// 24x v_wmma_f32_16x16x4_f32 (2 M-tiles x 4 N-tiles x 3 K-steps),
// fused BN+ReLU+max-over-points, transposed scatter into BEV.
// ---------------------------------------------------------------------------
#define FEAT_LDS (8 * 32 * 12)          // 3072 floats
#define WPAD_LDS (64 * 12)              // 768 floats

__global__ __launch_bounds__(256) void pillar_main(const float4* __restrict__ voxels,
                                                   const int* __restrict__ nump,
                                                   const int4* __restrict__ coords,
                                                   const float* __restrict__ W,
                                                   const float* __restrict__ scale,
                                                   const float* __restrict__ shift,
                                                   float* __restrict__ out, int P) {
  __shared__ float smem[FEAT_LDS + WPAD_LDS];   // 15 KB
  float* wlds = smem + FEAT_LDS;

  int tid = threadIdx.x;
  int wave = tid >> 5;
  int lane = tid & 31;
  int p = blockIdx.x * 8 + wave;
  bool active = p < P;
  int pc = active ? p : 0;              // clamp; keeps EXEC all-ones for WMMA

  // ---- stage zero-padded W[64][12] once per block (no predicated loads) ----
#pragma unroll
  for (int i = tid; i < 640; i += 256)
    wlds[(i / 10) * 12 + (i % 10)] = W[i];
  if (tid < 64) {
    wlds[tid * 12 + 10] = 0.f;
    wlds[tid * 12 + 11] = 0.f;
  }

  // ---- per-point features ----
  float4 v = voxels[(long)pc * NPT_ + lane];
  float sx = wave_sum(v.x);
  float sy = wave_sum(v.y);
  float sz = wave_sum(v.z);
  int np = nump[pc];
  int4 cc = coords[pc];                 // (b, z, y, x)

  float f[12];
  compute_feat(v.x, v.y, v.z, v.w, sx, sy, sz, (float)np,
               cc.y, cc.z, cc.w, lane < np, f);
  f[10] = 0.f; f[11] = 0.f;             // K padding 10 -> 12

  // stage A-matrix rows in LDS (row stride 12 floats = 48B, 16B aligned)
  float* row = smem + wave * 384 + lane * 12;
  v4f w0 = {f[0], f[1], f[2],  f[3]};
  v4f w1 = {f[4], f[5], f[6],  f[7]};
  v4f w2 = {f[8], f[9], f[10], f[11]};
  ((v4f*)row)[0] = w0;
  ((v4f*)row)[1] = w1;
  ((v4f*)row)[2] = w2;
  __syncthreads();

  const float* base = smem + wave * 384;
  int lrow  = lane & 15;
  int khalf = lane >> 4;                // 0: k0..k0+1, 1: k0+2..k0+3

  long cell    = (long)cc.y + (long)cc.z * NX_ + (long)cc.w;  // z + y*NX + x
  long outbase = ((long)cc.x * 64) * GRID_ + cell;            // [B,64,GRID]

#pragma unroll
  for (int nt = 0; nt < 4; ++nt) {
    v8f acc0 = {0.f, 0.f, 0.f, 0.f, 0.f, 0.f, 0.f, 0.f};
    v8f acc1 = {0.f, 0.f, 0.f, 0.f, 0.f, 0.f, 0.f, 0.f};
    int n = nt * 16 + lrow;
#pragma unroll
    for (int ks = 0; ks < 3; ++ks) {
      int kk = ks * 4 + khalf * 2;
      // A fragments (16x4 f32): points 0-15 and 16-31
      v2f a0, a1, bf;
      a0.x = base[lrow * 12 + kk];
      a0.y = base[lrow * 12 + kk + 1];
      a1.x = base[(lrow + 16) * 12 + kk];
      a1.y = base[(lrow + 16) * 12 + kk + 1];
      // B fragment (4x16 f32): B[k][n] = Wpad[n][k], zero-padded past K=10
      bf.x = wlds[n * 12 + kk];
      bf.y = wlds[n * 12 + kk + 1];
      acc0 = __builtin_amdgcn_wmma_f32_16x16x4_f32(false, a0, false, bf,
                                                   (short)0, acc0, false, false);
      acc1 = __builtin_amdgcn_wmma_f32_16x16x4_f32(false, a1, false, bf,
                                                   (short)0, acc1, false, false);
    }
    // fused BN + ReLU + max over 32 points
    int cch = nt * 16 + lrow;
    float a = scale[cch];
    float b = shift[cch];
    float mx = 0.f;                      // ReLU output >= 0
#pragma unroll
    for (int r = 0; r < 8; ++r) {
      float y0 = fmaxf(a * acc0[r] + b, 0.f);
      float y1 = fmaxf(a * acc1[r] + b, 0.f);
      mx = fmaxf(mx, fmaxf(y0, y1));
    }
    mx = fmaxf(mx, __shfl_xor(mx, 16, 32));  // combine row halves
    if (active && lane < 16)
      out[outbase + (long)cch * GRID_] = mx;
  }
}

// ---------------------------------------------------------------------------
extern "C" void kernel_launch(void* const* d_in, const int* in_sizes, int n_in,
                              void* d_out, int out_size, void* d_ws, size_t ws_size,
                              hipStream_t stream) {
  const float* voxels = (const float*)d_in[0];
  const int*   nump   = (const int*)d_in[1];
  const int*   coords = (const int*)d_in[2];
  const float* W      = (const float*)d_in[3];
  const float* gamma  = (const float*)d_in[4];
  const float* beta   = (const float*)d_in[5];
  float* out = (float*)d_out;
  float* ws  = (float*)d_ws;
  int P = in_sizes[1];

  // 1) zero-fill 438.8 MB BEV output (bandwidth-dominant step)
  zero_out_k<<<4096, 256, 0, stream>>>(out, (long)out_size);
  // 2) zero stat accumulators
  zero_ws_k<<<1, 128, 0, stream>>>(ws);
  // 3) accumulate S (10) and M (55) over all masked feature rows
  pillar_stats<<<512, 256, 0, stream>>>((const float4*)voxels, nump,
                                        (const int4*)coords, ws, P);
  // 4) per-channel BN scale/shift
  pillar_finalize<<<1, 64, 0, stream>>>(W, gamma, beta, ws,
                                        1.0f / ((float)P * (float)NPT_));
  // 5) WMMA GEMM + BN + ReLU + max + scatter (one pillar per wave)
  int blocks = (P + 7) / 8;
  pillar_main<<<blocks, 256, 0, stream>>>((const float4*)voxels, nump,
                                          (const int4*)coords, W,
                                          ws + 128, ws + 192, out, P);
}